// RNN_71786083385630
// MI455X (gfx1250) — compile-verified
//
#include <hip/hip_runtime.h>
#include <math.h>

// ---------------------------------------------------------------------------
// Fused GRU sequence model for MI455X (gfx1250, wave32).
//
// Kernel 1 (scan): 128 workgroups x 2 waves; each WG owns a 16-row batch tile.
//   Wave 0 (producer): preprocess-u/x + gates_x (h-independent) -> LDS ring.
//   Wave 1 (consumer): gh + GRU cell (the true serial chain), writes h_seq,
//                      decodes only the final step.
// Kernel 2 (decoder): batch-parallel over (L-1)*B rows, reads h from h_seq.
//
// All matmuls: V_WMMA_F32_16X16X4_F32 (full fp32).  Transcendentals:
// hardware v_tanh_f32 when available, branch-free exp/rcp fallback otherwise.
// ---------------------------------------------------------------------------

typedef __attribute__((ext_vector_type(2))) float v2f;
typedef __attribute__((ext_vector_type(8))) float v8f;

#define L_SEQ 1024
#define B_SZ  2048
#define IN_F  4
#define OBS_F 4
#define K_F   16

static __device__ __forceinline__ v8f wmma4(v2f a, v2f b, v8f c) {
  // D(16x16,f32) = A(16x4,f32) * B(4x16,f32) + C
  return __builtin_amdgcn_wmma_f32_16x16x4_f32(false, a, false, b, (short)0, c,
                                               false, false);
}

static __device__ __forceinline__ v8f splat8(float v) {
  v8f c;
#pragma unroll
  for (int i = 0; i < 8; ++i) c[i] = v;
  return c;
}

#if __has_builtin(__builtin_amdgcn_tanhf)
#define HAS_HW_TANH 1
static __device__ __forceinline__ float fast_tanh(float x) {
  return __builtin_amdgcn_tanhf(x);   // gfx1250 v_tanh_f32
}
#else
static __device__ __forceinline__ float fast_tanh(float x) {
  // branch-free: e = exp(-2|x|) in (0,1], tanh|x| = (1-e)/(1+e), fast rcp
  float ax = __builtin_fabsf(x);
  float e  = __expf(-2.0f * ax);
  float t  = (1.0f - e) * __builtin_amdgcn_rcpf(1.0f + e);
  return __builtin_copysignf(t, x);
}
#endif

static __device__ __forceinline__ float sigm(float x) {
#ifdef HAS_HW_TANH
  return __builtin_fmaf(0.5f, fast_tanh(0.5f * x), 0.5f);
#else
  return __builtin_amdgcn_rcpf(1.0f + __expf(-x));
#endif
}

static __device__ __forceinline__ void tanh8(v8f& a) {
#pragma unroll
  for (int i = 0; i < 8; ++i) a[i] = fast_tanh(a[i]);
}

static __device__ __forceinline__ void wbar() {
  __builtin_amdgcn_wave_barrier();  // compiler fence; LDS is in-order per wave
}

// B fragment of W^T for out = X @ W^T.  W is [Nrows, Kdim] row-major.
// B(4x16) layout: v0 = {K=kb (lanes 0-15), K=kb+2 (lanes 16-31)},
//                 v1 = {K=kb+1,            K=kb+3}          (ISA 7.12.2)
static __device__ __forceinline__ v2f bfrag(const float* __restrict__ W,
                                            int Nrows, int Kdim, int ntile,
                                            int kt, int lane) {
  int n  = ntile * 16 + (lane & 15);
  int kb = kt * 4 + ((lane >> 4) << 1);
  v2f r;
  if (n < Nrows) {
    r.x = W[n * Kdim + kb];
    r.y = W[n * Kdim + kb + 1];
  } else {
    r.x = 0.0f;
    r.y = 0.0f;
  }
  return r;
}

// A fragment (16x4) for K-step kt from a row-major LDS tile (stride floats).
static __device__ __forceinline__ v2f afrag_lds(const float* buf, int stride,
                                                int kt, int lane) {
  int m  = lane & 15;
  int kb = kt * 4 + ((lane >> 4) << 1);
  v2f r;
  r.x = buf[m * stride + kb];
  r.y = buf[m * stride + kb + 1];
  return r;
}

// Store a C/D tile (16x16) to a row-major LDS tile at column offset c0.
// C layout: vgpr r holds row r (lanes 0-15) and row r+8 (lanes 16-31).
static __device__ __forceinline__ void cstore_lds(float* buf, int stride,
                                                  int c0, v8f acc, int lane) {
  int col = c0 + (lane & 15);
  int rb  = (lane >> 4) * 8;
#pragma unroll
  for (int r = 0; r < 8; ++r) buf[(rb + r) * stride + col] = acc[r];
}

// ---------------------------------------------------------------------------
// Kernel 1: pipelined GRU scan (producer/consumer, 2 waves per workgroup)
// ---------------------------------------------------------------------------
__global__ __launch_bounds__(64) void gru_scan_kernel(
    const float* __restrict__ useq, const float* __restrict__ oseq,
    const float* __restrict__ puW1, const float* __restrict__ pub1,
    const float* __restrict__ puW2, const float* __restrict__ pub2,
    const float* __restrict__ pxW1, const float* __restrict__ pxb1,
    const float* __restrict__ pxW2, const float* __restrict__ pxb2,
    const float* __restrict__ Wih, const float* __restrict__ bih,
    const float* __restrict__ Whh, const float* __restrict__ bhh,
    const float* __restrict__ dW1, const float* __restrict__ db1,
    const float* __restrict__ dW2, const float* __restrict__ db2,
    float* __restrict__ xout, float* __restrict__ hout) {
  __shared__ __attribute__((aligned(32))) float tmp16[16 * 16];  // producer
  __shared__ __attribute__((aligned(32))) float buf32[16 * 32];  // producer+tail
  __shared__ __attribute__((aligned(32))) float hbuf[16 * 16];   // consumer
  __shared__ __attribute__((aligned(32))) float gbuf[2 * 3 * 256];  // ring

  const int lane  = threadIdx.x & 31;
  const int wave  = threadIdx.x >> 5;
  const int b0    = blockIdx.x * 16;
  const int m     = lane & 15;
  const int khalf = (lane >> 4) << 1;
  const int rb    = (lane >> 4) * 8;
  const bool is_prod = (wave == 0);

  // role-specific weight fragments (wave-uniform branches)
  v2f Bpu1, Bpx1, Bpu2[4], Bpx2[4], Bih_[3][8];       // producer
  v2f Bhh_[3][4];                                     // consumer
  float vb_pu1 = 0, vb_pu2 = 0, vb_px1 = 0, vb_px2 = 0;
  float vb_ir = 0, vb_iz = 0, vb_in = 0;
  float vb_hr = 0, vb_hz = 0, vb_hn = 0;
  v8f h = splat8(0.0f);

  if (is_prod) {
    Bpu1 = bfrag(puW1, 16, 4, 0, 0, lane);
    Bpx1 = bfrag(pxW1, 16, 4, 0, 0, lane);
#pragma unroll
    for (int kt = 0; kt < 4; ++kt) {
      Bpu2[kt] = bfrag(puW2, 16, 16, 0, kt, lane);
      Bpx2[kt] = bfrag(pxW2, 16, 16, 0, kt, lane);
    }
#pragma unroll
    for (int kt = 0; kt < 8; ++kt)
#pragma unroll
      for (int g = 0; g < 3; ++g) Bih_[g][kt] = bfrag(Wih, 48, 32, g, kt, lane);
    vb_pu1 = pub1[m]; vb_pu2 = pub2[m];
    vb_px1 = pxb1[m]; vb_px2 = pxb2[m];
    vb_ir = bih[m]; vb_iz = bih[16 + m]; vb_in = bih[32 + m];
  } else {
#pragma unroll
    for (int kt = 0; kt < 4; ++kt)
#pragma unroll
      for (int g = 0; g < 3; ++g) Bhh_[g][kt] = bfrag(Whh, 48, 16, g, kt, lane);
    vb_hr = bhh[m]; vb_hz = bhh[16 + m]; vb_hn = bhh[32 + m];
    cstore_lds(hbuf, 16, 0, h, lane);   // h_{-1} = 0
    wbar();
  }

  for (int t = 0; t < L_SEQ; ++t) {
    const size_t rowbase = (size_t)t * B_SZ + b0;

    if (is_prod) {
      // ---- preprocess u ----
      v2f au = *(const v2f*)(useq + (rowbase + m) * IN_F + khalf);
      v8f acc = wmma4(au, Bpu1, splat8(vb_pu1));
      tanh8(acc);
      cstore_lds(tmp16, 16, 0, acc, lane);
      wbar();
      v8f e = splat8(vb_pu2);
#pragma unroll
      for (int kt = 0; kt < 4; ++kt)
        e = wmma4(afrag_lds(tmp16, 16, kt, lane), Bpu2[kt], e);
      tanh8(e);
      wbar();
      cstore_lds(buf32, 32, 0, e, lane);   // inp[:,0:16]
      // ---- preprocess x ----
      v2f ax = *(const v2f*)(oseq + (rowbase + m) * OBS_F + khalf);
      acc = wmma4(ax, Bpx1, splat8(vb_px1));
      tanh8(acc);
      wbar();
      cstore_lds(tmp16, 16, 0, acc, lane);
      wbar();
      e = splat8(vb_px2);
#pragma unroll
      for (int kt = 0; kt < 4; ++kt)
        e = wmma4(afrag_lds(tmp16, 16, kt, lane), Bpx2[kt], e);
      tanh8(e);
      cstore_lds(buf32, 32, 16, e, lane);  // inp[:,16:32]
      wbar();
      // ---- gates_x = inp @ Wih^T + bih ----
      v8f gr = splat8(vb_ir), gz = splat8(vb_iz), gn = splat8(vb_in);
#pragma unroll
      for (int kt = 0; kt < 8; ++kt) {
        v2f a2 = afrag_lds(buf32, 32, kt, lane);
        gr = wmma4(a2, Bih_[0][kt], gr);
        gz = wmma4(a2, Bih_[1][kt], gz);
        gn = wmma4(a2, Bih_[2][kt], gn);
      }
      float* slot = gbuf + (t & 1) * 768;   // C-layout pass-through
      *(v8f*)(slot + lane * 8)       = gr;
      *(v8f*)(slot + 256 + lane * 8) = gz;
      *(v8f*)(slot + 512 + lane * 8) = gn;
    }

    __syncthreads();  // publish gates[t]; producer proceeds to t+1 meanwhile

    if (!is_prod) {
      const float* slot = gbuf + (t & 1) * 768;
      v8f gr = *(const v8f*)(slot + lane * 8);
      v8f gz = *(const v8f*)(slot + 256 + lane * 8);
      v8f gn = *(const v8f*)(slot + 512 + lane * 8);
      // h_seq[t] = h_{t-1}
#pragma unroll
      for (int r = 0; r < 8; ++r) hout[(rowbase + rb + r) * K_F + m] = h[r];
      // gh = h @ Whh^T + bhh
      v8f hr = splat8(vb_hr), hz = splat8(vb_hz), hn = splat8(vb_hn);
#pragma unroll
      for (int kt = 0; kt < 4; ++kt) {
        v2f ah = afrag_lds(hbuf, 16, kt, lane);
        hr = wmma4(ah, Bhh_[0][kt], hr);
        hz = wmma4(ah, Bhh_[1][kt], hz);
        hn = wmma4(ah, Bhh_[2][kt], hn);
      }
      // GRU cell
#pragma unroll
      for (int r = 0; r < 8; ++r) {
        float rg = sigm(gr[r] + hr[r]);
        float zg = sigm(gz[r] + hz[r]);
        float ng = fast_tanh(gn[r] + rg * hn[r]);
        h[r] = (1.0f - zg) * ng + zg * h[r];
      }
      wbar();
      cstore_lds(hbuf, 16, 0, h, lane);
      wbar();
    }
  }

  // tail: consumer decodes the final step (its h is never stored globally)
  if (!is_prod) {
    v2f Bd1_[2][4], Bd2_[8];
#pragma unroll
    for (int kt = 0; kt < 4; ++kt) {
      Bd1_[0][kt] = bfrag(dW1, 32, 16, 0, kt, lane);
      Bd1_[1][kt] = bfrag(dW1, 32, 16, 1, kt, lane);
    }
#pragma unroll
    for (int kt = 0; kt < 8; ++kt) Bd2_[kt] = bfrag(dW2, 4, 32, 0, kt, lane);
    const float vb_d1a = db1[m], vb_d1b = db1[16 + m];
    const float vb_d2 = (m < 4) ? db2[m] : 0.0f;

    v8f d0 = splat8(vb_d1a), d1 = splat8(vb_d1b);
#pragma unroll
    for (int kt = 0; kt < 4; ++kt) {
      v2f ah = afrag_lds(hbuf, 16, kt, lane);
      d0 = wmma4(ah, Bd1_[0][kt], d0);
      d1 = wmma4(ah, Bd1_[1][kt], d1);
    }
    tanh8(d0);
    tanh8(d1);
    wbar();
    cstore_lds(buf32, 32, 0, d0, lane);
    cstore_lds(buf32, 32, 16, d1, lane);
    wbar();
    v8f xo = splat8(vb_d2);
#pragma unroll
    for (int kt = 0; kt < 8; ++kt)
      xo = wmma4(afrag_lds(buf32, 32, kt, lane), Bd2_[kt], xo);
    if (m < 4) {
      const size_t rowb = (size_t)(L_SEQ - 1) * B_SZ + b0;
#pragma unroll
      for (int r = 0; r < 8; ++r)
        xout[(rowb + rb + r) * OBS_F + m] = xo[r];
    }
  }
}

// ---------------------------------------------------------------------------
// Kernel 2: batch-parallel decoder for t = 0..L-2 (h_all[t] = h_seq[t+1])
// ---------------------------------------------------------------------------
__global__ __launch_bounds__(256) void decoder_kernel(
    const float* __restrict__ hin, const float* __restrict__ dW1,
    const float* __restrict__ db1, const float* __restrict__ dW2,
    const float* __restrict__ db2, float* __restrict__ xout) {
  __shared__ __attribute__((aligned(32))) float dbuf[8][16 * 32];

  const int lane  = threadIdx.x & 31;
  const int wave  = threadIdx.x >> 5;
  const int m     = lane & 15;
  const int khalf = (lane >> 4) << 1;
  const int rb    = (lane >> 4) * 8;
  float* buf = dbuf[wave];

  const long long TOT = (long long)(L_SEQ - 1) * B_SZ;
  const long long R   = ((long long)blockIdx.x * 8 + wave) * 16;  // x-row base
  if (R >= TOT) return;  // wave-uniform

  v2f Bd1_[2][4], Bd2_[8];
#pragma unroll
  for (int kt = 0; kt < 4; ++kt) {
    Bd1_[0][kt] = bfrag(dW1, 32, 16, 0, kt, lane);
    Bd1_[1][kt] = bfrag(dW1, 32, 16, 1, kt, lane);
  }
#pragma unroll
  for (int kt = 0; kt < 8; ++kt) Bd2_[kt] = bfrag(dW2, 4, 32, 0, kt, lane);
  const float vb_d1a = db1[m], vb_d1b = db1[16 + m];
  const float vb_d2 = (m < 4) ? db2[m] : 0.0f;

  const float* hrow = hin + (size_t)(R + B_SZ) * K_F;  // h_seq[t+1]
  v8f d0 = splat8(vb_d1a), d1 = splat8(vb_d1b);
#pragma unroll
  for (int kt = 0; kt < 4; ++kt) {
    v2f ah = *(const v2f*)(hrow + m * K_F + kt * 4 + khalf);
    d0 = wmma4(ah, Bd1_[0][kt], d0);
    d1 = wmma4(ah, Bd1_[1][kt], d1);
  }
  tanh8(d0);
  tanh8(d1);
  cstore_lds(buf, 32, 0, d0, lane);
  cstore_lds(buf, 32, 16, d1, lane);
  wbar();
  v8f xo = splat8(vb_d2);
#pragma unroll
  for (int kt = 0; kt < 8; ++kt)
    xo = wmma4(afrag_lds(buf, 32, kt, lane), Bd2_[kt], xo);
  if (m < 4) {
#pragma unroll
    for (int r = 0; r < 8; ++r)
      xout[(R + rb + r) * OBS_F + m] = xo[r];
  }
}

extern "C" void kernel_launch(void* const* d_in, const int* in_sizes, int n_in,
                              void* d_out, int out_size, void* d_ws,
                              size_t ws_size, hipStream_t stream) {
  (void)in_sizes; (void)n_in; (void)out_size; (void)d_ws; (void)ws_size;
  const float* useq = (const float*)d_in[0];
  const float* oseq = (const float*)d_in[1];
  const float* puW1 = (const float*)d_in[2];
  const float* pub1 = (const float*)d_in[3];
  const float* puW2 = (const float*)d_in[4];
  const float* pub2 = (const float*)d_in[5];
  const float* pxW1 = (const float*)d_in[6];
  const float* pxb1 = (const float*)d_in[7];
  const float* pxW2 = (const float*)d_in[8];
  const float* pxb2 = (const float*)d_in[9];
  const float* Wih  = (const float*)d_in[10];
  const float* bih  = (const float*)d_in[11];
  const float* Whh  = (const float*)d_in[12];
  const float* bhh  = (const float*)d_in[13];
  const float* dW1  = (const float*)d_in[14];
  const float* db1  = (const float*)d_in[15];
  const float* dW2  = (const float*)d_in[16];
  const float* db2  = (const float*)d_in[17];

  float* out  = (float*)d_out;
  float* xout = out;                                       // [L,B,OBS]
  float* hout = out + (size_t)L_SEQ * B_SZ * OBS_F;        // [L,B,K]

  // 1) pipelined scan: 128 tiles, 2 waves each
  hipLaunchKernelGGL(gru_scan_kernel, dim3(B_SZ / 16), dim3(64), 0, stream,
                     useq, oseq, puW1, pub1, puW2, pub2, pxW1, pxb1, pxW2,
                     pxb2, Wih, bih, Whh, bhh, dW1, db1, dW2, db2, xout, hout);

  // 2) batch-parallel decoder over t = 0..L-2 (reads h_seq just written)
  const long long tot_rows = (long long)(L_SEQ - 1) * B_SZ;  // 2095104
  const int nblocks = (int)((tot_rows / 16 + 7) / 8);        // 8 waves/block
  hipLaunchKernelGGL(decoder_kernel, dim3(nblocks), dim3(256), 0, stream,
                     hout, dW1, db1, dW2, db2, xout);
}